// TreeSelfAttention_30116310680239
// MI455X (gfx1250) — compile-verified
//
#include <hip/hip_runtime.h>

// TreeSelfAttention for MI455X (gfx1250), fp32 throughout.
// B=8, T=4096, D=1024, N=15 nodes, tree depth 4.
// Memory-bound op (~440 MiB total traffic); GEMMs (0.5 GFLOP) done with
// V_WMMA_F32_16X16X4_F32 for exact fp32 accumulation.

typedef __attribute__((ext_vector_type(2))) float v2f;
typedef __attribute__((ext_vector_type(8))) float v8f;

#define D_DIM   1024
#define B_DIM   8
#define N_NODES 15

// ---------------------------------------------------------------------------
// Kernel 1: leaves = mean over 256 timesteps; only first 8 leaves are used.
// grid (4 dchunks, 8 leaves, 8 batch), block 256.
// ---------------------------------------------------------------------------
__global__ __launch_bounds__(256)
void leaves_kernel(const float* __restrict__ x, float* __restrict__ vec)
{
    const int d = blockIdx.x * 256 + threadIdx.x;   // 0..1023
    const int n = blockIdx.y;                       // leaf 0..7
    const int b = blockIdx.z;                       // batch 0..7
    const float* base = x + ((size_t)b * 4096 + (size_t)n * 256) * D_DIM + d;
    float s = 0.0f;
    #pragma unroll 8
    for (int t = 0; t < 256; ++t) s += base[(size_t)t * D_DIM];
    vec[((size_t)b * 8 + n) * D_DIM + d] = s * (1.0f / 256.0f);
}

// ---------------------------------------------------------------------------
// Kernel 2: one tree level. Per node: proj = A(16x2048) * Wp[node](2048x1024),
// rows 0..7 of A are batch, rows 8..15 zero-padded (masked at fragment load).
// One workgroup = (node, 128-col chunk); 8 waves, one 16x16 tile per wave.
// A staged in LDS (8 rows x 2048, XOR-swizzled, exactly 64 KB).
//
// WMMA f32 16x16x4 fragment layouts (ISA 7.12.2):
//   A: lane l holds row m=l%16; vgpr v holds K = 2*(l/16)+v
//   B: lane l holds col n=l%16; vgpr v holds K = 2*(l/16)+v
//   C: vgpr r, lane l -> M = r + 8*(l/16), N = l%16
// ---------------------------------------------------------------------------
__global__ __launch_bounds__(256)
void tree_level_kernel(const float* __restrict__ inBuf, int inNodeStride,
                       int childBase, int dup,
                       float* __restrict__ active,
                       const float* __restrict__ Wp,
                       const float* __restrict__ bp,
                       const float* __restrict__ bt,
                       int startNode)
{
    __shared__ float ldsA[8 * 2048];   // 64 KB, XOR-swizzled columns

    const int node  = blockIdx.x >> 3;     // node within level
    const int chunk = blockIdx.x & 7;      // 128-column chunk
    const int tid   = threadIdx.x;
    const int lane  = tid & 31;
    const int wave  = tid >> 5;

    const int leftIdx  = dup ? node : (childBase + 2 * node);
    const int rightIdx = dup ? node : (childBase + 2 * node + 1);

    // Stage A rows 0..7 (the real batch rows) into LDS.
    // swizzle: column k stored at k ^ (row*8) -> conflict-free strided reads.
    for (int idx = tid; idx < 8 * 2048; idx += 256) {
        const int m = idx >> 11;          // row 0..7
        const int k = idx & 2047;         // col 0..2047
        const int src = (k < 1024)
            ? ((m * inNodeStride + leftIdx)  * D_DIM + k)
            : ((m * inNodeStride + rightIdx) * D_DIM + (k - 1024));
        ldsA[(m << 11) + (k ^ (m << 3))] = inBuf[src];
    }
    __syncthreads();

    const int ncol  = chunk * 128 + wave * 16;   // N-tile base column
    const int nodeG = startNode + node;          // global node index
    const int m     = lane & 15;                 // fragment row / col index
    const int g     = lane >> 4;                 // lane half
    const bool keep = (lane & 8) == 0;           // rows 8..15 are zero pad
    const int  mrow = m & 7;
    const int  swz  = mrow << 3;

    const float* wpBase = Wp + (size_t)nodeG * 2048 * D_DIM
                             + (size_t)(2 * g) * D_DIM + (ncol + m);
    const float* aRow   = &ldsA[mrow << 11];

    v8f c = {};
    #pragma unroll 4
    for (int kb = 0; kb < 2048; kb += 4) {
        const int kk = kb + 2 * g;                       // even -> 8B aligned
        v2f a = *(const v2f*)(aRow + (kk ^ swz));        // K=kk, kk+1
        a.x = keep ? a.x : 0.0f;
        a.y = keep ? a.y : 0.0f;
        v2f b;
        b.x = wpBase[(size_t)kb * D_DIM];                // K = kb + 2g
        b.y = wpBase[(size_t)kb * D_DIM + D_DIM];        // K = kb + 2g + 1
        c = __builtin_amdgcn_wmma_f32_16x16x4_f32(
                false, a, false, b, (short)0, c, false, false);
    }

    // Epilogue: bias + hard gate (forward value of _spike is (u>0)), write
    // rows M=0..7 (= batch, held in lanes 0..15) into active[b][nodeG][d].
    const int d     = ncol + m;
    const float bias = bp[nodeG * D_DIM + d];
    const float thr  = bt[nodeG * D_DIM + d];
    #pragma unroll
    for (int r = 0; r < 8; ++r) {
        float proj  = c[r] + bias;
        float gated = (proj - thr) > 0.0f ? proj : 0.0f;
        if (lane < 16)
            active[((size_t)r * N_NODES + nodeG) * D_DIM + d] = gated;
    }
}

// ---------------------------------------------------------------------------
// Kernel 3: weights = softmax(node_w over N); mixture[b,d] = sum_n w*active.
// grid 8 blocks x 128 threads -> one thread per d column.
// ---------------------------------------------------------------------------
__global__ __launch_bounds__(128)
void mixture_kernel(const float* __restrict__ node_w,
                    const float* __restrict__ active,
                    float* __restrict__ mixture)
{
    const int d = blockIdx.x * 128 + threadIdx.x;
    float w[N_NODES];
    float mx = -3.402823466e38f;
    #pragma unroll
    for (int n = 0; n < N_NODES; ++n) {
        w[n] = node_w[n * D_DIM + d];
        mx = fmaxf(mx, w[n]);
    }
    float denom = 0.0f;
    #pragma unroll
    for (int n = 0; n < N_NODES; ++n) {
        w[n] = expf(w[n] - mx);
        denom += w[n];
    }
    const float inv = 1.0f / denom;
    for (int b = 0; b < B_DIM; ++b) {
        float acc = 0.0f;
        #pragma unroll
        for (int n = 0; n < N_NODES; ++n)
            acc += w[n] * active[((size_t)b * N_NODES + n) * D_DIM + d];
        mixture[b * D_DIM + d] = acc * inv;
    }
}

// ---------------------------------------------------------------------------
// Kernel 4: out = LayerNorm(x + mixture[b]) * gamma + beta.
// One wave32 per (b,t) row of 1024; float4 vector traffic; wave reduction.
// This is the bandwidth kernel: 128 MiB read + 128 MiB write.
// ---------------------------------------------------------------------------
__global__ __launch_bounds__(256)
void add_ln_kernel(const float* __restrict__ x,
                   const float* __restrict__ mixture,
                   const float* __restrict__ gamma,
                   const float* __restrict__ beta,
                   float* __restrict__ out)
{
    const int lane = threadIdx.x & 31;
    const int wave = threadIdx.x >> 5;
    const size_t row = (size_t)blockIdx.x * 8 + wave;   // 0..32767
    const int b = (int)(row >> 12);                     // 4096 rows per batch

    const float4* xr = (const float4*)(x + row * D_DIM);
    const float4* mr = (const float4*)(mixture + (size_t)b * D_DIM);

    float4 v[8];
    float sum = 0.0f, sumsq = 0.0f;
    #pragma unroll
    for (int j = 0; j < 8; ++j) {
        const int idx = j * 32 + lane;
        float4 xv = xr[idx];
        float4 mv = mr[idx];
        float4 t;
        t.x = xv.x + mv.x; t.y = xv.y + mv.y;
        t.z = xv.z + mv.z; t.w = xv.w + mv.w;
        v[j] = t;
        sum   += t.x + t.y + t.z + t.w;
        sumsq += t.x * t.x + t.y * t.y + t.z * t.z + t.w * t.w;
    }
    #pragma unroll
    for (int off = 16; off > 0; off >>= 1) {
        sum   += __shfl_xor(sum,   off, 32);
        sumsq += __shfl_xor(sumsq, off, 32);
    }
    const float mean = sum * (1.0f / 1024.0f);
    const float var  = sumsq * (1.0f / 1024.0f) - mean * mean;
    const float rstd = rsqrtf(var + 1e-5f);

    float4* outr = (float4*)(out + row * D_DIM);
    const float4* gr = (const float4*)gamma;
    const float4* br = (const float4*)beta;
    #pragma unroll
    for (int j = 0; j < 8; ++j) {
        const int idx = j * 32 + lane;
        float4 g = gr[idx], be = br[idx], t = v[j];
        float4 o;
        o.x = (t.x - mean) * rstd * g.x + be.x;
        o.y = (t.y - mean) * rstd * g.y + be.y;
        o.z = (t.z - mean) * rstd * g.z + be.z;
        o.w = (t.w - mean) * rstd * g.w + be.w;
        outr[idx] = o;
    }
}

// ---------------------------------------------------------------------------
// Host launcher. Workspace: vec (256 KB) + active (480 KB) + mixture (32 KB).
// ---------------------------------------------------------------------------
extern "C" void kernel_launch(void* const* d_in, const int* in_sizes, int n_in,
                              void* d_out, int out_size, void* d_ws, size_t ws_size,
                              hipStream_t stream) {
    (void)in_sizes; (void)n_in; (void)out_size; (void)ws_size;
    const float* x      = (const float*)d_in[0];
    const float* Wp     = (const float*)d_in[1];
    const float* bp     = (const float*)d_in[2];
    const float* bt     = (const float*)d_in[3];
    // d_in[4] (aw) is multiplied by 0.0 in the reference -> unused.
    const float* node_w = (const float*)d_in[5];
    const float* gamma  = (const float*)d_in[6];
    const float* beta   = (const float*)d_in[7];
    float* out = (float*)d_out;

    float* vec     = (float*)d_ws;                         // 8*8*1024
    float* active  = vec + 8 * 8 * D_DIM;                  // 8*15*1024
    float* mixture = active + 8 * N_NODES * D_DIM;         // 8*1024

    leaves_kernel<<<dim3(4, 8, 8), 256, 0, stream>>>(x, vec);

    // level 3: 8 nodes (global 7..14), input = vec duplicated [vec,vec]
    tree_level_kernel<<<dim3(8 * 8), 256, 0, stream>>>(
        vec, 8, 0, 1, active, Wp, bp, bt, 7);
    // level 2: 4 nodes (global 3..6), children = global 7..14
    tree_level_kernel<<<dim3(4 * 8), 256, 0, stream>>>(
        active, N_NODES, 7, 0, active, Wp, bp, bt, 3);
    // level 1: 2 nodes (global 1..2), children = global 3..6
    tree_level_kernel<<<dim3(2 * 8), 256, 0, stream>>>(
        active, N_NODES, 3, 0, active, Wp, bp, bt, 1);
    // level 0: 1 node (global 0), children = global 1..2
    tree_level_kernel<<<dim3(1 * 8), 256, 0, stream>>>(
        active, N_NODES, 1, 0, active, Wp, bp, bt, 0);

    mixture_kernel<<<dim3(8), 128, 0, stream>>>(node_w, active, mixture);

    add_ln_kernel<<<dim3(4096), 256, 0, stream>>>(x, mixture, gamma, beta, out);
}